// SimpleMPNN_14431090114818
// MI455X (gfx1250) — compile-verified
//
#include <hip/hip_runtime.h>
#include <hip/hip_bf16.h>

// GCN on MI455X (gfx1250, wave32).
// - fp32 WMMA (v_wmma_f32_16x16x4_f32) for the N x 64 x 64 per-layer GEMMs;
//   weight (B) fragments hoisted into registers and reused across 4 row tiles
//   per wave (64 WMMAs per wave, weights loaded once).
// - Edge gather/scatter is L2-resident (xw/agg = 25.6MB each << 192MB L2);
//   one wave per edge, each lane moves a float2 (coalesced 256B row) with
//   global_atomic_add_f32 scatter.
// - Self-loop contribution folded into bias+ReLU pass (no atomics needed).

#define DFEAT 64
#define MTILES_PER_WAVE 4

typedef float v2f __attribute__((ext_vector_type(2)));
typedef float v8f __attribute__((ext_vector_type(8)));

// ---------------- degree / normalization ----------------

__global__ void degree_count_kernel(const int* __restrict__ col,
                                    float* __restrict__ deg, int nE) {
    int i = blockIdx.x * blockDim.x + threadIdx.x;
    if (i < nE) atomicAdd(&deg[col[i]], 1.0f);
}

__global__ void make_dinv_kernel(float* __restrict__ deg, int n) {
    int i = blockIdx.x * blockDim.x + threadIdx.x;
    if (i < n) {
        // +1 accounts for the self loop; deg >= 1 always.
        deg[i] = rsqrtf(deg[i] + 1.0f);
    }
}

// ---------------- fp32 WMMA GEMM: out[N,64] = H[N,64] @ W[64,64] ----------------
// blockDim = 128 (4 waves). Wave w owns cols [16w, 16w+16) and processes
// MTILES_PER_WAVE consecutive 16-row tiles, reusing its register-resident
// weight fragments across all of them.
//
// A (16x4 f32) per-lane layout: lanes 0-15 hold K=0,1 ; lanes 16-31 hold K=2,3
//   -> one float2 load per lane per step.
// B (4x16 f32): VGPR0 = rows K=0 (lanes 0-15) / K=2 (lanes 16-31), VGPR1 = K=1/K=3.
// D (16x16 f32, 8 VGPRs): VGPR v -> row m0+v (lanes 0-15) / m0+v+8 (lanes 16-31).

__global__ void __launch_bounds__(128)
gemm64_wmma_kernel(const float* __restrict__ H, const float* __restrict__ W,
                   float* __restrict__ out, int n_rows, int mtiles_total) {
    const int lane = threadIdx.x & 31;
    const int wave = threadIdx.x >> 5;
    const int half = lane >> 4;     // 0 for lanes 0-15, 1 for lanes 16-31
    const int l16  = lane & 15;
    const int n    = wave * 16 + l16;

    // Hoist all 16 weight (B) fragments into registers once: 32 VGPRs.
    v2f bf[16];
#pragma unroll
    for (int kk = 0; kk < 16; ++kk) {
        const int k = kk * 4 + 2 * half;
        bf[kk].x = W[(size_t)k * DFEAT + n];
        bf[kk].y = W[(size_t)(k + 1) * DFEAT + n];
    }

#pragma unroll
    for (int mt = 0; mt < MTILES_PER_WAVE; ++mt) {
        int tile = blockIdx.x * MTILES_PER_WAVE + mt;
        if (tile > mtiles_total - 1) tile = mtiles_total - 1;  // clamp tail; EXEC stays full
        const int m0 = tile * 16;

        int mA = m0 + l16;
        if (mA > n_rows - 1) mA = n_rows - 1;
        const float* __restrict__ hrow = H + (size_t)mA * DFEAT;

        v8f c = {};
#pragma unroll
        for (int kk = 0; kk < 16; ++kk) {
            const int k = kk * 4 + 2 * half;
            v2f a;
            a.x = hrow[k];
            a.y = hrow[k + 1];
            c = __builtin_amdgcn_wmma_f32_16x16x4_f32(
                /*neg_a=*/false, a, /*neg_b=*/false, bf[kk],
                /*c_mod=*/(short)0, c, /*reuse_a=*/false, /*reuse_b=*/false);
        }

#pragma unroll
        for (int v = 0; v < 8; ++v) {
            const int mr = m0 + v + 8 * half;
            if (mr < n_rows) out[(size_t)mr * DFEAT + n] = c[v];
        }
    }
}

// ---------------- edge gather * norm -> scatter-add ----------------
// One wave per edge (grid-stride over waves). Lane l moves floats [2l, 2l+1]
// of the 64-float row: a fully coalesced 256B transaction per edge.

__global__ void edge_scatter_kernel(const int* __restrict__ row,
                                    const int* __restrict__ col,
                                    const float* __restrict__ dinv,
                                    const float* __restrict__ xw,
                                    float* __restrict__ agg, int nE) {
    const int lane = threadIdx.x & 31;
    const int wid  = (blockIdx.x * blockDim.x + threadIdx.x) >> 5;
    const int nw   = (gridDim.x * blockDim.x) >> 5;
    for (int e = wid; e < nE; e += nw) {
        const int r = row[e];
        const int t = col[e];
        const float nrm = dinv[r] * dinv[t];
        const float2 v = ((const float2*)(xw + (size_t)r * DFEAT))[lane];
        float* dst = agg + (size_t)t * DFEAT + 2 * lane;
        atomicAdd(dst + 0, v.x * nrm);
        atomicAdd(dst + 1, v.y * nrm);
    }
}

// ---------------- self-loop + bias + ReLU ----------------
// h[i,d] = relu(agg[i,d] + xw[i,d]*dinv[i]^2 + b[d])

__global__ void bias_relu_kernel(const float* __restrict__ agg,
                                 const float* __restrict__ xw,
                                 const float* __restrict__ dinv,
                                 const float* __restrict__ b,
                                 float* __restrict__ h, int n) {
    const long long i = (long long)blockIdx.x * blockDim.x + threadIdx.x;
    const long long total = (long long)n * DFEAT;
    if (i < total) {
        const int node = (int)(i >> 6);
        const int d = (int)(i & (DFEAT - 1));
        const float di = dinv[node];
        const float v = agg[i] + xw[i] * di * di + b[d];
        h[i] = v > 0.0f ? v : 0.0f;
    }
}

// ---------------- final FC: out[i] = h[i,:] . Wfc + bfc ----------------
// One wave per node; lane l handles float2 [2l,2l+1]; shfl_xor tree reduce.

__global__ void fc_kernel(const float* __restrict__ h,
                          const float* __restrict__ wfc,
                          const float* __restrict__ bfc,
                          float* __restrict__ out, int n) {
    const int lane = threadIdx.x & 31;
    const int wid  = (blockIdx.x * blockDim.x + threadIdx.x) >> 5;
    const int nw   = (gridDim.x * blockDim.x) >> 5;
    const float2 wv = ((const float2*)wfc)[lane];
    const float bias = bfc[0];
    for (int i = wid; i < n; i += nw) {
        const float2 hv = ((const float2*)(h + (size_t)i * DFEAT))[lane];
        float v = hv.x * wv.x + hv.y * wv.y;
#pragma unroll
        for (int off = 16; off > 0; off >>= 1) v += __shfl_xor(v, off, 32);
        if (lane == 0) out[i] = v + bias;
    }
}

// ---------------- launcher ----------------

extern "C" void kernel_launch(void* const* d_in, const int* in_sizes, int n_in,
                              void* d_out, int out_size, void* d_ws, size_t ws_size,
                              hipStream_t stream) {
    const int N = in_sizes[0] / DFEAT;   // 100000
    const int E = in_sizes[1] / 2;       // 1600000

    const float* x   = (const float*)d_in[0];
    const int*   ei  = (const int*)d_in[1];
    const int*   row = ei;       // edge_index[0] = source
    const int*   col = ei + E;   // edge_index[1] = target
    const float* Wl[4] = {(const float*)d_in[2], (const float*)d_in[4],
                          (const float*)d_in[6], (const float*)d_in[8]};
    const float* Bl[4] = {(const float*)d_in[3], (const float*)d_in[5],
                          (const float*)d_in[7], (const float*)d_in[9]};
    const float* Wfc = (const float*)d_in[10];
    const float* bfc = (const float*)d_in[11];

    float* ws   = (float*)d_ws;
    float* dinv = ws;                                // N floats
    float* xw   = dinv + N;                          // N*64
    float* agg  = xw + (size_t)N * DFEAT;            // N*64
    float* hbuf = agg + (size_t)N * DFEAT;           // N*64

    // normalization coefficients (recomputed every call; deterministic)
    hipMemsetAsync(dinv, 0, (size_t)N * sizeof(float), stream);
    degree_count_kernel<<<(E + 255) / 256, 256, 0, stream>>>(col, dinv, E);
    make_dinv_kernel<<<(N + 255) / 256, 256, 0, stream>>>(dinv, N);

    const int mtiles = (N + 15) / 16;
    const int gemm_blocks = (mtiles + MTILES_PER_WAVE - 1) / MTILES_PER_WAVE;
    const long long nd = (long long)N * DFEAT;
    const int br_blocks = (int)((nd + 255) / 256);

    const float* hin = x;
    for (int l = 0; l < 4; ++l) {
        gemm64_wmma_kernel<<<gemm_blocks, 128, 0, stream>>>(hin, Wl[l], xw, N, mtiles);
        hipMemsetAsync(agg, 0, nd * sizeof(float), stream);
        edge_scatter_kernel<<<16384, 256, 0, stream>>>(row, col, dinv, xw, agg, E);
        bias_relu_kernel<<<br_blocks, 256, 0, stream>>>(agg, xw, dinv, Bl[l], hbuf, N);
        hin = hbuf;
    }

    fc_kernel<<<4096, 256, 0, stream>>>(hin, Wfc, bfc, (float*)d_out, N);
}